// ValueIterNet_85160611545491
// MI455X (gfx1250) — compile-verified
//
#include <hip/hip_runtime.h>
#include <stdint.h>

#define AS1 __attribute__((address_space(1)))
#define AS3 __attribute__((address_space(3)))

typedef int v4i __attribute__((ext_vector_type(4)));

namespace {
constexpr int N  = 40;
constexpr int NP = N * N * N * N;   // 2,560,000 grid points
constexpr int A  = 5;
constexpr int TILE = 256;           // points per tile == blockDim.x
constexpr int TILES_PER_BLOCK = 10; // 10,000 tiles total -> 1000 blocks
constexpr int NBLOCKS = NP / (TILE * TILES_PER_BLOCK);
constexpr int S1 = N * N * N, S2 = N * N, S3 = N;

// ---- CDNA5 async global->LDS copies (ASYNCcnt-tracked) ----------------------
__device__ __forceinline__ void async_copy_b128(const float* g, float* l) {
#if __has_builtin(__builtin_amdgcn_global_load_async_to_lds_b128)
  __builtin_amdgcn_global_load_async_to_lds_b128((AS1 v4i*)(AS1 void*)g,
                                                 (AS3 v4i*)(AS3 void*)l,
                                                 /*offset=*/0, /*cpol=*/0);
#else
  uint32_t lds_off = (uint32_t)(uintptr_t)l;
  asm volatile("global_load_async_to_lds_b128 %0, %1, off"
               :: "v"(lds_off), "v"(g) : "memory");
#endif
}

__device__ __forceinline__ void async_copy_b32(const float* g, float* l) {
#if __has_builtin(__builtin_amdgcn_global_load_async_to_lds_b32)
  __builtin_amdgcn_global_load_async_to_lds_b32((AS1 int*)(AS1 void*)g,
                                                (AS3 int*)(AS3 void*)l,
                                                /*offset=*/0, /*cpol=*/0);
#else
  uint32_t lds_off = (uint32_t)(uintptr_t)l;
  asm volatile("global_load_async_to_lds_b32 %0, %1, off"
               :: "v"(lds_off), "v"(g) : "memory");
#endif
}

template <int CNT>
__device__ __forceinline__ void wait_asynccnt() {
#if __has_builtin(__builtin_amdgcn_s_wait_asynccnt)
  __builtin_amdgcn_s_wait_asynccnt(CNT);
#else
  asm volatile("s_wait_asynccnt %0" :: "i"(CNT) : "memory");
#endif
}

// Stage one tile: 5 regions x 5120 B. Per region each lane issues one B128
// (lanes cover the first 4096 B) plus one B32 (last 1024 B) -> 2 async
// instructions per region, 10 per tile, uniform EXEC across all lanes so the
// per-wave ASYNCcnt is deterministic for the double-buffer wait.
__device__ __forceinline__ void load_tile(float (*buf)[TILE * A],
                                          const float* __restrict__ dsdt,
                                          const float* __restrict__ cost,
                                          int tileStart, int tid) {
#pragma unroll
  for (int r = 0; r < 5; ++r) {
    const float* src = (r < 4) ? dsdt + ((size_t)r * NP + tileStart) * A
                               : cost + (size_t)tileStart * A;
    async_copy_b128(src + tid * 4, &buf[r][tid * 4]);      // bytes [0, 4096)
    async_copy_b32(src + 1024 + tid, &buf[r][1024 + tid]); // bytes [4096, 5120)
  }
}
}  // namespace

__global__ __launch_bounds__(TILE)
void ValueIterNet_85160611545491_kernel(const float* __restrict__ J,
                                        const float* __restrict__ dsdt,
                                        const float* __restrict__ cost,
                                        float* __restrict__ out_v,
                                        int* __restrict__ out_a) {
  // double-buffered staging: [buf][region][point*A + a], 50 KB
  __shared__ float smem[2][5][TILE * A];

  const int tid   = threadIdx.x;
  const int tile0 = blockIdx.x * TILES_PER_BLOCK;

  load_tile(smem[0], dsdt, cost, tile0 * TILE, tid);  // prologue

  for (int t = 0; t < TILES_PER_BLOCK; ++t) {
    const int cur = t & 1;
    if (t + 1 < TILES_PER_BLOCK) {
      // previous compute pass on buffer cur^1 finished at the trailing barrier
      load_tile(smem[cur ^ 1], dsdt, cost, (tile0 + t + 1) * TILE, tid);
      wait_asynccnt<10>();  // tile t done (in-order), tile t+1 still in flight
    } else {
      wait_asynccnt<0>();
    }
    __syncthreads();  // all waves' portions of tile t visible in LDS

    // ---------------- compute: upwind partials of J --------------------------
    const int idx = (tile0 + t) * TILE + tid;
    int q = idx;
    const int i4 = q % N; q /= N;
    const int i3 = q % N; q /= N;
    const int i2 = q % N; const int i1 = q / N;

    const float Jc = J[idx];
    // dims 0,1 periodic: edge copies opposite-edge one-sided diff (per reference)
    float up0 = (i1 < N - 1) ? (J[idx + S1] - Jc)
                             : (J[idx - (N - 2) * S1] - J[idx - (N - 1) * S1]);
    float dn0 = (i1 > 0)     ? (Jc - J[idx - S1])
                             : (J[idx + (N - 1) * S1] - J[idx + (N - 2) * S1]);
    float up1 = (i2 < N - 1) ? (J[idx + S2] - Jc)
                             : (J[idx - (N - 2) * S2] - J[idx - (N - 1) * S2]);
    float dn1 = (i2 > 0)     ? (Jc - J[idx - S2])
                             : (J[idx + (N - 1) * S2] - J[idx + (N - 2) * S2]);
    // dims 2,3 non-periodic: zeros at the closed edge
    float up2 = (i3 < N - 1) ? (J[idx + S3] - Jc) : 0.0f;
    float dn2 = (i3 > 0)     ? (Jc - J[idx - S3]) : 0.0f;
    float up3 = (i4 < N - 1) ? (J[idx + 1] - Jc) : 0.0f;
    float dn3 = (i4 > 0)     ? (Jc - J[idx - 1]) : 0.0f;

    constexpr float c01 = 0.5f / 0.157f, c2 = 0.5f / 0.3f, c3 = 0.5f / 0.5f;
    up0 *= c01; dn0 *= c01; up1 *= c01; dn1 *= c01;
    up2 *= c2;  dn2 *= c2;  up3 *= c3;  dn3 *= c3;

    // LDS reads: stride-5 floats across lanes -> conflict-free (gcd(5,64)=1)
    const float* w0 = &smem[cur][0][tid * A];
    const float* w1 = &smem[cur][1][tid * A];
    const float* w2 = &smem[cur][2][tid * A];
    const float* w3 = &smem[cur][3][tid * A];
    const float* cc = &smem[cur][4][tid * A];

    float best = __builtin_inff();
    int   barg = 0;
#pragma unroll
    for (int a = 0; a < A; ++a) {
      float x0 = w0[a], x1 = w1[a], x2 = w2[a], x3 = w3[a];
      float t0 = up0 * fmaxf(x0, 0.0f) + dn0 * fminf(x0, 0.0f);
      float t1 = up1 * fmaxf(x1, 0.0f) + dn1 * fminf(x1, 0.0f);
      float t2 = up2 * fmaxf(x2, 0.0f) + dn2 * fminf(x2, 0.0f);
      float t3 = up3 * fmaxf(x3, 0.0f) + dn3 * fminf(x3, 0.0f);
      float v  = ((t0 + t1) + t2) + t3 + cc[a];  // GAMMA == 1.0
      if (v < best) { best = v; barg = a; }       // first-min tie-break
    }

    out_v[idx] = best;
    out_a[idx] = barg;

    __syncthreads();  // buffer `cur` must be fully read before t+2's async issue
  }
}

extern "C" void kernel_launch(void* const* d_in, const int* in_sizes, int n_in,
                              void* d_out, int out_size, void* d_ws, size_t ws_size,
                              hipStream_t stream) {
  const float* J    = (const float*)d_in[0];
  const float* dsdt = (const float*)d_in[1];
  const float* cost = (const float*)d_in[2];
  float* out_v = (float*)d_out;            // dJdt: first NP floats
  int*   out_a = (int*)d_out + NP;         // argmin: next NP as int32
  hipLaunchKernelGGL(ValueIterNet_85160611545491_kernel,
                     dim3(NBLOCKS), dim3(TILE), 0, stream,
                     J, dsdt, cost, out_v, out_a);
}